// mrla_base_layer_14216341750458
// MI455X (gfx1250) — compile-verified
//
#include <hip/hip_runtime.h>
#include <hip/hip_bf16.h>

typedef float v2f __attribute__((ext_vector_type(2)));
typedef float v8f __attribute__((ext_vector_type(8)));

#define Bn 16
#define Cn 512
#define HWn 784      // 28*28
#define HW4 196      // 784/4
#define Tn 16
#define Gn 16
#define Dn 32

// ---------------- Kernel 1: global average pool  y[b,c] = mean_{h,w} x ----------------
__global__ void k_pool(const float* __restrict__ x, float* __restrict__ y) {
    __shared__ float sd[256];
    const int bc  = blockIdx.x;          // 0..8191  (b*512+c)
    const int tid = threadIdx.x;
    const float4* xp = (const float4*)(x + bc * HWn);
    float s = 0.f;
    if (tid < HW4) { float4 v = xp[tid]; s = v.x + v.y + v.z + v.w; }
    sd[tid] = s;
    __syncthreads();
    for (int st = 128; st > 0; st >>= 1) {
        if (tid < st) sd[tid] += sd[tid + st];
        __syncthreads();
    }
    if (tid == 0) y[bc] = sd[0] * (1.0f / 784.0f);
}

// ------- Kernel 2: 5-tap channel convs (Q, k), assemble K cache = [prev_K ; k] -------
__global__ void k_qk(const float* __restrict__ y, const float* __restrict__ wq,
                     const float* __restrict__ wk, const float* __restrict__ prevK,
                     float* __restrict__ Qws, float* __restrict__ Kout) {
    const int i = blockIdx.x * 256 + threadIdx.x;   // grid covers 122880
    if (i < Bn * 15 * Cn) {                          // copy prev_K rows into K[b,t<15,c]
        const int c = i & (Cn - 1);
        const int t = (i >> 9) % 15;
        const int b = i / (Cn * 15);
        Kout[(b * Tn + t) * Cn + c] = prevK[i];
    }
    if (i < Bn * Cn) {                               // Q[b,c], k[b,c] -> K[b,15,c]
        const int b = i >> 9, c = i & (Cn - 1);
        const float* yb = y + b * Cn;
        float q = 0.f, kk = 0.f;
#pragma unroll
        for (int j = 0; j < 5; ++j) {
            const int cc = c - 2 + j;
            const float yv = (cc >= 0 && cc < Cn) ? yb[cc] : 0.f;
            q  += yv * wq[j];
            kk += yv * wk[j];
        }
        Qws[i] = q;
        Kout[(b * Tn + 15) * Cn + c] = kk;
    }
}

// ------- Kernel 3: depthwise 3x3 conv (zero pad), written straight into V[b,15,c,:,:] -------
__global__ void k_dwconv(const float* __restrict__ x, const float* __restrict__ wv,
                         float* __restrict__ Vout) {
    __shared__ float tile[30 * 30];
    const int bc = blockIdx.x;           // b*512+c
    const int b  = bc >> 9;
    const int c  = bc & (Cn - 1);
    const int tid = threadIdx.x;
    const float* xp = x + bc * HWn;
    for (int idx = tid; idx < 900; idx += 256) {
        const int th = idx / 30, tw = idx - th * 30;
        const int gh = th - 1,  gw = tw - 1;
        float v = 0.f;
        if (gh >= 0 && gh < 28 && gw >= 0 && gw < 28) v = xp[gh * 28 + gw];
        tile[idx] = v;
    }
    __syncthreads();
    const float* wc = wv + c * 9;
    const float w0 = wc[0], w1 = wc[1], w2 = wc[2];
    const float w3 = wc[3], w4 = wc[4], w5 = wc[5];
    const float w6 = wc[6], w7 = wc[7], w8 = wc[8];
    float* vp = Vout + ((b * Tn + 15) * Cn + c) * HWn;
    for (int pos = tid; pos < HWn; pos += 256) {
        const int h = pos / 28, w = pos - h * 28;
        const float* t0 = &tile[h * 30 + w];
        float acc = w0 * t0[0]  + w1 * t0[1]  + w2 * t0[2]
                  + w3 * t0[30] + w4 * t0[31] + w5 * t0[32]
                  + w6 * t0[60] + w7 * t0[61] + w8 * t0[62];
        vp[pos] = acc;
    }
}

// ------- Kernel 4: attention scores via V_WMMA_F32_16X16X4_F32 + shuffle softmax -------
// One wave32 per (b,g). GEMM mapping: C[m,t] = sum_d A[m,d]*B[d,t],
//   A[m,d] = Q[b,g,d]   (broadcast over the 16 M rows; ISA 16x4 f32 A layout:
//                        lane L: m=L%16, VGPR0 holds K = (L<16?0:2), VGPR1 = +1)
//   B[d,t] = K[b,t,g,d] (4x16 B: VGPR0 lanes0-15 -> K=0, lanes16-31 -> K=2; VGPR1 -> K=1/3)
// 8 chained WMMAs cover d=32; every lane ends up with S[t = lane%16] in acc[0].
__global__ void k_scores(const float* __restrict__ Q, const float* __restrict__ Kc,
                         float* __restrict__ att) {
    const int bb   = blockIdx.x >> 4;
    const int g    = blockIdx.x & 15;
    const int lane = threadIdx.x;        // 0..31, all active (EXEC must be all 1s)
    const int t    = lane & 15;
    const int hi   = (lane >> 4) << 1;   // 0 or 2: which K-pair this half-wave holds
    const float* qp = Q  + bb * Cn + g * Dn;
    const float* kp = Kc + (bb * Tn + t) * Cn + g * Dn;
    v8f acc = {};
#pragma unroll
    for (int ch = 0; ch < 8; ++ch) {
        const int dbase = ch * 4;
        v2f a; a.x = qp[dbase + hi];  a.y = qp[dbase + hi + 1];
        v2f bm; bm.x = kp[dbase + hi]; bm.y = kp[dbase + hi + 1];
        acc = __builtin_amdgcn_wmma_f32_16x16x4_f32(
                  false, a, false, bm, (short)0, acc, false, false);
    }
    float S = acc[0] * 0.17677669529663687f;   // 1/sqrt(32)
    // softmax over t (16 lanes per half-wave; masks 1,2,4,8 stay inside each group)
    float m = S;
#pragma unroll
    for (int off = 8; off >= 1; off >>= 1) m = fmaxf(m, __shfl_xor(m, off, 32));
    const float e = __expf(S - m);
    float s = e;
#pragma unroll
    for (int off = 8; off >= 1; off >>= 1) s += __shfl_xor(s, off, 32);
    if (lane < 16) att[(bb * Gn + g) * Tn + lane] = e / s;
}

// ------- Kernel 5: fused  out = sum_t a[t]*V[t]  +  prev_V -> V cache copy (float4) -------
__global__ void k_fuse(const float4* __restrict__ prevV, const float* __restrict__ att,
                       float4* __restrict__ Vout, float4* __restrict__ out) {
    const int i = blockIdx.x * 256 + threadIdx.x;    // 0 .. 16*512*196-1 (exact grid)
    const int p = i % HW4;
    const int c = (i / HW4) & (Cn - 1);
    const int b = i / (HW4 * Cn);
    const int g = c >> 5;
    const float* ap = att + (b * Gn + g) * Tn;
    float a[16];
#pragma unroll
    for (int t = 0; t < 16; ++t) a[t] = ap[t];

    // t = 15 slice was written by the depthwise conv (already in place in Vout)
    float4 v15 = Vout[((b * Tn + 15) * Cn + c) * HW4 + p];
    float4 acc;
    acc.x = a[15] * v15.x; acc.y = a[15] * v15.y;
    acc.z = a[15] * v15.z; acc.w = a[15] * v15.w;
#pragma unroll
    for (int t = 0; t < 15; ++t) {
        const float4 pv = prevV[((b * 15 + t) * Cn + c) * HW4 + p];
        Vout[((b * Tn + t) * Cn + c) * HW4 + p] = pv;   // cache copy (read once, write once)
        acc.x += a[t] * pv.x; acc.y += a[t] * pv.y;
        acc.z += a[t] * pv.z; acc.w += a[t] * pv.w;
    }
    out[(b * Cn + c) * HW4 + p] = acc;
}

extern "C" void kernel_launch(void* const* d_in, const int* in_sizes, int n_in,
                              void* d_out, int out_size, void* d_ws, size_t ws_size,
                              hipStream_t stream) {
    const float* x     = (const float*)d_in[0];
    const float* prevK = (const float*)d_in[1];
    const float* prevV = (const float*)d_in[2];
    const float* wq    = (const float*)d_in[3];
    const float* wk    = (const float*)d_in[4];
    const float* wv    = (const float*)d_in[5];

    float* out  = (float*)d_out;                 // [16,512,28,28]
    float* Kout = out + Bn * Cn * HWn;           // [16,16,512]      @ 6,422,528
    float* Vout = Kout + Bn * Tn * Cn;           // [16,16,512,28,28] @ 6,553,600

    float* ws     = (float*)d_ws;
    float* ws_y   = ws;                          // 8192 floats
    float* ws_q   = ws + 8192;                   // 8192 floats
    float* ws_att = ws + 16384;                  // 4096 floats  (total 80 KB)

    k_pool  <<<Bn * Cn, 256, 0, stream>>>(x, ws_y);
    k_qk    <<<(Bn * 15 * Cn) / 256, 256, 0, stream>>>(ws_y, wq, wk, prevK, ws_q, Kout);
    k_dwconv<<<Bn * Cn, 256, 0, stream>>>(x, wv, Vout);
    k_scores<<<Bn * Gn, 32, 0, stream>>>(ws_q, Kout, ws_att);
    k_fuse  <<<(Bn * Cn * HW4) / 256, 256, 0, stream>>>((const float4*)prevV, ws_att,
                                                        (float4*)Vout, (float4*)out);
}